// FedGATConv_11639361372435
// MI455X (gfx1250) — compile-verified
//
#include <hip/hip_runtime.h>
#include <hip/hip_bf16.h>

typedef __attribute__((ext_vector_type(2))) float v2f;
typedef __attribute__((ext_vector_type(4))) float v4f;
typedef __attribute__((ext_vector_type(8))) float v8f;

#define N_NODES 4096
#define F_IN    128
#define F_OUT   128
#define NPOLY   9            // P = MAX_DEG + 1
#define S_COLS  16
#define QDIM    (NPOLY * S_COLS)   // 144
#define NPB     16           // nodes per block
#define THREADS 256          // 8 waves (wave32)

__global__ __launch_bounds__(THREADS)
void fedgat_fused_kernel(const float* __restrict__ M1,
                         const float* __restrict__ M2,
                         const float* __restrict__ K1,
                         const float* __restrict__ K2,
                         const float* __restrict__ att1,
                         const float* __restrict__ att2,
                         const float* __restrict__ weight,
                         const float* __restrict__ poly,
                         float* __restrict__ out)
{
    __shared__ float lds_Dp[NPB][QDIM];    // 9216 B : poly-weighted powers
    __shared__ float lds_E [NPB][F_IN];    // 8192 B : E_pre tile (A matrix)
    __shared__ float lds_Fp[THREADS];      // 1024 B : Fden partials
    __shared__ float lds_F [NPB];          //   64 B : Fden per node

    const int t    = threadIdx.x;
    const int base = blockIdx.x * NPB;     // first node of this block
    const int nn   = t >> 4;               // node-in-block 0..15
    const int s    = t & 15;               // aggregation column 0..15
    const int node = base + nn;

    // ---- Step 1: D[nn][s] = att1 . M1[node,:,s] + att2 . M2[node,:,s] ----
    const float* m1 = M1 + (size_t)node * F_IN * S_COLS + s;
    const float* m2 = M2 + (size_t)node * F_IN * S_COLS + s;
    float d = 0.0f;
    #pragma unroll 8
    for (int f = 0; f < F_IN; ++f) {       // att1[f]/att2[f] are uniform -> SMEM loads
        d = fmaf(att1[f], m1[f * S_COLS], d);
        d = fmaf(att2[f], m2[f * S_COLS], d);
    }

    // ---- Step 2: Dp[nn][p*16+s] = poly[p] * d^p ; fused Fden partial ----
    const float* k2 = K2 + (size_t)node * QDIM + s;
    float dpow  = 1.0f;
    float fpart = 0.0f;
    #pragma unroll
    for (int p = 0; p < NPOLY; ++p) {
        float v = poly[p] * dpow;
        lds_Dp[nn][p * S_COLS + s] = v;
        fpart = fmaf(v, k2[p * S_COLS], fpart);
        dpow *= d;
    }
    lds_Fp[t] = fpart;
    __syncthreads();

    if (t < NPB) {                          // Fden[n] = sum over 16 columns
        float acc = 0.0f;
        #pragma unroll
        for (int i = 0; i < S_COLS; ++i) acc += lds_Fp[t * S_COLS + i];
        lds_F[t] = acc;
    }

    // ---- Step 3: stream K1 once: E_pre[nn][f] = sum_q Dp[nn][q]*K1[node,q,f] ----
    const int wave = t >> 5;                // 0..7
    const int lane = t & 31;
    #pragma unroll
    for (int i = 0; i < 2; ++i) {           // each wave owns 2 nodes
        const int mn = wave * 2 + i;
        const float* k1 = K1 + (size_t)(base + mn) * QDIM * F_IN + lane * 4;
        v4f acc = {0.0f, 0.0f, 0.0f, 0.0f};
        for (int q = 0; q < QDIM; ++q) {    // b128 load per lane, 512B per wave
            const float dq = lds_Dp[mn][q]; // LDS broadcast
            const v4f kv = *(const v4f*)(k1 + (size_t)q * F_IN);
            acc.x = fmaf(dq, kv.x, acc.x);
            acc.y = fmaf(dq, kv.y, acc.y);
            acc.z = fmaf(dq, kv.z, acc.z);
            acc.w = fmaf(dq, kv.w, acc.w);
        }
        *(v4f*)&lds_E[mn][lane * 4] = acc;
    }
    __syncthreads();

    // ---- Step 4: C[16 nodes, 16 cols] = E_pre(16x128) @ weight(128x16) via WMMA ----
    // wave w -> output columns [w*16, w*16+16); f32 16x16x4, 32 K-steps.
    // 32-bit A layout (ISA 7.12.2): v0: K=k0 (lanes 0-15) / K=k0+2 (lanes 16-31); v1: +1.
    const int col  = wave * 16 + (lane & 15);
    const int koff = (lane < 16) ? 0 : 2;
    const int mrow = lane & 15;
    v8f c = {};
    #pragma unroll
    for (int kk = 0; kk < F_IN / 4; ++kk) {
        const int k0 = kk * 4 + koff;
        v2f a = *(const v2f*)&lds_E[mrow][k0];              // 8B aligned ds_load
        v2f b;
        b.x = weight[(size_t)k0 * F_OUT + col];             // B mirrors A striping
        b.y = weight[(size_t)(k0 + 1) * F_OUT + col];
        c = __builtin_amdgcn_wmma_f32_16x16x4_f32(
                /*neg_a=*/false, a, /*neg_b=*/false, b,
                /*c_mod=*/(short)0, c, /*reuse_a=*/false, /*reuse_b=*/false);
    }

    // ---- Step 5: normalize by Fden and store (C/D layout: M = r + 8*(lane>=16)) ----
    #pragma unroll
    for (int r = 0; r < 8; ++r) {
        const int m = r + ((lane < 16) ? 0 : 8);
        out[(size_t)(base + m) * F_OUT + col] = c[r] / lds_F[m];
    }
}

extern "C" void kernel_launch(void* const* d_in, const int* in_sizes, int n_in,
                              void* d_out, int out_size, void* d_ws, size_t ws_size,
                              hipStream_t stream) {
    const float* M1     = (const float*)d_in[0];
    const float* M2     = (const float*)d_in[1];
    const float* K1     = (const float*)d_in[2];
    const float* K2     = (const float*)d_in[3];
    const float* att1   = (const float*)d_in[4];
    const float* att2   = (const float*)d_in[5];
    const float* weight = (const float*)d_in[6];
    const float* poly   = (const float*)d_in[7];
    float* outp = (float*)d_out;

    dim3 grid(N_NODES / NPB);   // 256 blocks x 16 nodes
    fedgat_fused_kernel<<<grid, THREADS, 0, stream>>>(
        M1, M2, K1, K2, att1, att2, weight, poly, outp);
}